// EpipolarPropagation_5652176961973
// MI455X (gfx1250) — compile-verified
//
#include <hip/hip_runtime.h>

// ---------------------------------------------------------------------------
// EpipolarPropagation for MI455X (gfx1250, wave32)
// B=8, C=16, H=W=512.  Memory-bound: ~280 MiB moved -> ~13us at 23.3 TB/s.
// WMMA (V_WMMA_F32_16X16X4_F32) computes the per-batch M = K @ R @ Kinv chain.
// ---------------------------------------------------------------------------

typedef __attribute__((ext_vector_type(2))) float v2f;
typedef __attribute__((ext_vector_type(8))) float v8f;

#define HW (512 * 512)

// One wave (32 threads) per batch. Computes M_b = K @ R_b @ Kinv via two
// 16x16x4 f32 WMMAs (3x3 matrices zero-padded into the fragments), and
// KT_b = K @ T_b.  Results: Mws[b*16 + 0..8] = M row-major, [9..11] = KT.
__global__ __launch_bounds__(32) void prep_kernel(const float* __restrict__ T,
                                                  const float* __restrict__ R,
                                                  const float* __restrict__ K,
                                                  const float* __restrict__ Kinv,
                                                  float* __restrict__ Mws) {
    const int b    = blockIdx.x;
    const int lane = threadIdx.x;
    const int m    = lane & 15;   // A-fragment row / B,D-fragment column
    const int hf   = lane >> 4;   // half select (K-pairs per ISA layout)
    const int n    = m;

    __shared__ float lds[64];

    const float* Rb = R + b * 9;
    const float* Tb = T + b * 3;

    auto padK  = [&](int r, int c) -> float { return (r < 3 && c < 3) ? K[r * 3 + c]    : 0.0f; };
    auto padR  = [&](int r, int c) -> float { return (r < 3 && c < 3) ? Rb[r * 3 + c]   : 0.0f; };
    auto padKi = [&](int r, int c) -> float { return (r < 3 && c < 3) ? Kinv[r * 3 + c] : 0.0f; };

    // A (16x4 f32): lanes 0-15 hold K-cols {0,1}, lanes 16-31 hold {2,3}
    v2f a1; a1.x = padK(m, 2 * hf); a1.y = padK(m, 2 * hf + 1);
    // B (4x16 f32): lane = N column; VGPR0/1 hold row pairs per half
    v2f b1; b1.x = padR(2 * hf, n); b1.y = padR(2 * hf + 1, n);
    v8f cz = {};
    // KR = K @ R  (padded 16x16)
    v8f d1 = __builtin_amdgcn_wmma_f32_16x16x4_f32(false, a1, false, b1,
                                                   (short)0, cz, false, false);

    // D layout: VGPR j, lanes 0-15 -> row j, lanes 16-31 -> row j+8.
    // Only rows 0..2 of KR are nonzero; bounce them through LDS to re-pack
    // into an A-fragment for the second multiply.
    if (hf == 0) {
        lds[0 * 16 + n] = d1[0];
        lds[1 * 16 + n] = d1[1];
        lds[2 * 16 + n] = d1[2];
    }
    __syncthreads();

    v2f a2;
    a2.x = (m < 3) ? lds[m * 16 + 2 * hf]     : 0.0f;
    a2.y = (m < 3) ? lds[m * 16 + 2 * hf + 1] : 0.0f;
    v2f b2; b2.x = padKi(2 * hf, n); b2.y = padKi(2 * hf + 1, n);
    // M = (K @ R) @ Kinv
    v8f d2 = __builtin_amdgcn_wmma_f32_16x16x4_f32(false, a2, false, b2,
                                                   (short)0, cz, false, false);

    float* Mb = Mws + b * 16;
    if (hf == 0 && n < 3) {
        Mb[0 * 3 + n] = d2[0];
        Mb[1 * 3 + n] = d2[1];
        Mb[2 * 3 + n] = d2[2];
    }
    if (lane == 0) {
        // KT = K @ T  (so y = K @ (T/depth) = KT / depth)
        for (int i = 0; i < 3; ++i)
            Mb[9 + i] = K[i * 3 + 0] * Tb[0] + K[i * 3 + 1] * Tb[1] + K[i * 3 + 2] * Tb[2];
    }
}

// idxbuf[B*HW] = -1 via wide 128-bit stores.
__global__ __launch_bounds__(256) void init_kernel(int4* __restrict__ idxbuf) {
    unsigned t = blockIdx.x * 256u + threadIdx.x;   // 524288 threads total
    idxbuf[t] = make_int4(-1, -1, -1, -1);
}

// Per (b, pixel p): compute warped target, scatter source index p with
// atomicMax (resolves duplicate scatters deterministically).
__global__ __launch_bounds__(256) void project_kernel(const float* __restrict__ depth,
                                                      const float* __restrict__ Mws,
                                                      int* __restrict__ idxbuf) {
    const int b = blockIdx.y;
    const int p = blockIdx.x * 256 + threadIdx.x;   // HW / 256 blocks in x
    const float* Mb = Mws + b * 16;                  // uniform -> scalar loads

    const float row = (float)(p >> 9);   // gx (ij meshgrid: row index)
    const float col = (float)(p & 511);  // gy

    float d = depth[b * HW + p];
    d = fmaxf(d, 0.1f);                  // MIN_DEPTH clamp
    const float rinv = 1.0f / d;

    float t0 = fmaf(Mb[0], row, fmaf(Mb[1], col, Mb[2])) + Mb[9]  * rinv;
    float t1 = fmaf(Mb[3], row, fmaf(Mb[4], col, Mb[5])) + Mb[10] * rinv;
    float t2 = fmaf(Mb[6], row, fmaf(Mb[7], col, Mb[8])) + Mb[11] * rinv;
    if (t0 == 0.0f) t0 = 1e-4f;
    if (t1 == 0.0f) t1 = 1e-4f;
    if (t2 == 0.0f) t2 = 1e-4f;

    const float inv2 = 1.0f / t2;
    const float u = fminf(fmaxf(t0 * inv2, 0.0f), 511.0f);
    const float v = fminf(fmaxf(t1 * inv2, 0.0f), 511.0f);
    const int flat = ((int)u << 9) + (int)v;         // t0*W + t1

    atomicMax(&idxbuf[b * HW + flat], p);
}

// Per (b, c, pooled cell): gather the 2x2 scatter-cell sources, take the max
// (untouched cell == 0.0, matching zero-initialized proj), and write the
// 2x2 nearest-upsampled block.  Fuses scatter-gather + maxpool + upsample.
__global__ __launch_bounds__(256) void pool_kernel(const float* __restrict__ image,
                                                   const int* __restrict__ idxbuf,
                                                   float* __restrict__ out) {
    const unsigned t = blockIdx.x * 256u + threadIdx.x;  // 8*16*256*256 threads
    const int px = t & 255;
    const int py = (t >> 8) & 255;
    const int c  = (t >> 16) & 15;
    const int b  = t >> 20;

    const int* ib = idxbuf + b * HW;
    const int base = (py * 2) * 512 + px * 2;
    const int2 i01 = *(const int2*)(ib + base);          // row 2*py
    const int2 i23 = *(const int2*)(ib + base + 512);    // row 2*py+1

    const float* img = image + (size_t)(b * 16 + c) * HW;   // L2-resident
    const float v0 = (i01.x >= 0) ? img[i01.x] : 0.0f;
    const float v1 = (i01.y >= 0) ? img[i01.y] : 0.0f;
    const float v2 = (i23.x >= 0) ? img[i23.x] : 0.0f;
    const float v3 = (i23.y >= 0) ? img[i23.y] : 0.0f;
    const float val = fmaxf(fmaxf(v0, v1), fmaxf(v2, v3));

    float* o = out + (size_t)(b * 16 + c) * HW;
    const float2 vv = make_float2(val, val);
    *(float2*)(o + base)       = vv;
    *(float2*)(o + base + 512) = vv;
}

extern "C" void kernel_launch(void* const* d_in, const int* in_sizes, int n_in,
                              void* d_out, int out_size, void* d_ws, size_t ws_size,
                              hipStream_t stream) {
    const float* image = (const float*)d_in[0];  // [8,16,512,512]
    const float* depth = (const float*)d_in[1];  // [8,512,512]
    const float* T     = (const float*)d_in[2];  // [8,3,1]
    const float* R     = (const float*)d_in[3];  // [8,3,3]
    const float* K     = (const float*)d_in[4];  // [3,3]
    const float* Kinv  = (const float*)d_in[5];  // [3,3]
    float* out = (float*)d_out;                  // [8,16,512,512] f32

    float* Mws  = (float*)d_ws;                  // 8 batches x 16 floats = 512 B
    int* idxbuf = (int*)((char*)d_ws + 512);     // 8 * HW ints = 8 MiB

    prep_kernel<<<8, 32, 0, stream>>>(T, R, K, Kinv, Mws);
    init_kernel<<<(8 * HW / 4) / 256, 256, 0, stream>>>((int4*)idxbuf);
    project_kernel<<<dim3(HW / 256, 8), 256, 0, stream>>>(depth, Mws, idxbuf);
    pool_kernel<<<(8 * 16 * 256 * 256) / 256, 256, 0, stream>>>(image, idxbuf, out);
}